// CVRPDecoder_40321152975398
// MI455X (gfx1250) — compile-verified
//
#include <hip/hip_runtime.h>
#include <math.h>

// Shapes from the reference
#define B 64
#define P 64
#define NN 1000
#define H 128

// LDS strides: even (8B-aligned v2f reads) and bank-conflict-free:
// row-indexed pair reads hit banks {4l+c,4l+c+1} (lanes 0-15) vs
// {4l+c+2,4l+c+3} (lanes 16-31) -> disjoint.
#define EMB_STR 132   // 64 x 132 floats (emb chunk / vsum A / final_q A)
#define MSK_STR 68    // 64 x 68 floats  (mask A)

typedef float v2f __attribute__((ext_vector_type(2)));
typedef float v8f __attribute__((ext_vector_type(8)));
typedef int   v4i __attribute__((ext_vector_type(4)));

// Full-precision matrix op: D(16x16,f32) = A(16x4,f32) * B(4x16,f32) + C
// A: lane(0..15) holds M=lane, K={k,k+1}; lane(16..31): M=lane-16, K={k+2,k+3}
// B: lane holds column N=lane&15, rows K = {2*(lane>>4)+0, 2*(lane>>4)+1}
// C/D: vgpr r -> M = r + 8*(lane>>4), N = lane&15
__device__ __forceinline__ v8f wmma_f32(v2f a, v2f b, v8f c) {
  return __builtin_amdgcn_wmma_f32_16x16x4_f32(false, a, false, b, (short)0, c,
                                               false, false);
}

// ---- CDNA5 async global->LDS staging (guarded; falls back to b128 copy) ----
#if __has_builtin(__builtin_amdgcn_global_load_async_to_lds_b128)
#define HAVE_ASYNC_LDS 1
#else
#define HAVE_ASYNC_LDS 0
#endif

__device__ __forceinline__ void copy16_g2l(float* l, const float* g) {
#if HAVE_ASYNC_LDS
  __builtin_amdgcn_global_load_async_to_lds_b128(
      (__attribute__((address_space(1))) v4i*)g,
      (__attribute__((address_space(3))) v4i*)l,
      0, 0);
#else
  *(float4*)l = *(const float4*)g;
#endif
}

__device__ __forceinline__ void wait_async_copies() {
#if HAVE_ASYNC_LDS
#if __has_builtin(__builtin_amdgcn_s_wait_asynccnt)
  __builtin_amdgcn_s_wait_asynccnt(0);
#else
  asm volatile("s_wait_asynccnt 0x0" ::: "memory");
#endif
#endif
}

// ---------------------------------------------------------------------------
// Kernel 1: final_q[b,p,h] = encq1@Wq_first + last_emb@Wq_last
//                          + (mask01@emb/N)@W_visited + q_graph + load_emb
// One workgroup (8 wave32) per batch b.
// ---------------------------------------------------------------------------
__global__ void __launch_bounds__(256)
fq_kernel(const float* __restrict__ emb, const float* __restrict__ encq1,
          const int* __restrict__ last_node, const float* __restrict__ loadv,
          const float* __restrict__ gmask,
          const float* __restrict__ Wq_graph, const float* __restrict__ Wq_first,
          const float* __restrict__ Wq_last, const float* __restrict__ Wv,
          const float* __restrict__ W_load, const float* __restrict__ b_load,
          float* __restrict__ fq)
{
  __shared__ __align__(16) float s_emb[64 * EMB_STR]; // 33.8 KB
  __shared__ __align__(16) float s_msk[64 * MSK_STR]; // 17.4 KB
  __shared__ float s_part[256];
  __shared__ float s_mean[H];
  __shared__ float s_qg[H];
  __shared__ float s_load[P];

  const int b    = blockIdx.x;
  const int tid  = threadIdx.x;
  const int lane = tid & 31;
  const int wid  = tid >> 5;
  const int lrow = lane & 15;         // M / N position within tile
  const int khi  = (lane >> 4) * 2;   // K offset for upper half-wave
  const float* embB = emb + (size_t)b * NN * H;

  // --- phase 1: column mean over N (all 256 threads), q_graph matvec -------
  {
    const int h = tid & 127, half = tid >> 7;
    float s = 0.f;
    for (int n = half; n < NN; n += 2) s += embB[(size_t)n * H + h];
    s_part[tid] = s;
  }
  if (tid < P) s_load[tid] = loadv[b * P + tid];
  __syncthreads();
  if (tid < H) s_mean[tid] = (s_part[tid] + s_part[tid + 128]) * (1.0f / NN);
  __syncthreads();
  if (tid < H) {
    float s = 0.f;
    for (int k = 0; k < H; ++k) s += s_mean[k] * Wq_graph[k * H + tid];
    s_qg[tid] = s;
  }

  // --- phase 2: vsum[p][h] = sum_n mask01[p][n] * emb[n][h]  (WMMA, K=N) ---
  const int p0 = (wid >> 1) * 16;     // 4 M-tiles, 2 waves each
  const int hb = (wid & 1) * 64;      // each wave owns 64 H columns (4 tiles)
  v8f acc[4] = {};

  for (int n0 = 0; n0 < NN; n0 += 64) {
    const int rows = (NN - n0 < 64) ? (NN - n0) : 64;
    __syncthreads();
    // stage emb chunk rows [0,rows) via async global->LDS b128
    for (int i = tid; i < rows * 32; i += 256) {
      int r = i >> 5, c4 = (i & 31) * 4;
      copy16_g2l(&s_emb[r * EMB_STR + c4], embB + (size_t)(n0 + r) * H + c4);
    }
    for (int i = tid; i < (64 - rows) * 128; i += 256) {   // zero tail rows
      int r = rows + (i >> 7), c = i & 127;
      s_emb[r * EMB_STR + c] = 0.f;
    }
    for (int i = tid; i < 64 * 64; i += 256) {             // stage mask 0/1
      int p = i >> 6, k = i & 63, n = n0 + k;
      float g = (n < NN) ? gmask[((size_t)b * P + p) * NN + n] : 0.f;
      s_msk[p * MSK_STR + k] = (g < -1e30f) ? 1.f : 0.f;
    }
    wait_async_copies();
    __syncthreads();
    for (int kk = 0; kk < 64; kk += 4) {
      v2f a = *(const v2f*)&s_msk[(p0 + lrow) * MSK_STR + kk + khi];
#pragma unroll
      for (int t = 0; t < 4; ++t) {
        v2f bb;
        bb.x = s_emb[(kk + khi + 0) * EMB_STR + hb + 16 * t + lrow];
        bb.y = s_emb[(kk + khi + 1) * EMB_STR + hb + 16 * t + lrow];
        acc[t] = wmma_f32(a, bb, acc[t]);
      }
    }
  }
  __syncthreads();
  // park vsum/N back into LDS as an A matrix [p][h]
  const float invN = 1.0f / NN;
#pragma unroll
  for (int t = 0; t < 4; ++t)
#pragma unroll
    for (int r = 0; r < 8; ++r)
      s_emb[(p0 + r + 8 * (lane >> 4)) * EMB_STR + hb + 16 * t + lrow] =
          acc[t][r] * invN;
  __syncthreads();

  // --- phase 3: three chained K=128 WMMA GEMMs -----------------------------
  v8f f[4] = {};
  const int ln = last_node[b * P + p0 + lrow];
  const v2f* encRow  = (const v2f*)(encq1 + ((size_t)b * P + p0 + lrow) * H);
  const v2f* lastRow = (const v2f*)(embB + (size_t)ln * H);

  for (int k0 = 0; k0 < H; k0 += 4) {
    const int ki = (k0 >> 1) + (lane >> 4);   // v2f index = (k0+khi)/2
    v2f a3 = *(const v2f*)&s_emb[(p0 + lrow) * EMB_STR + k0 + khi];
    v2f a1 = encRow[ki];
    v2f a2 = lastRow[ki];
#pragma unroll
    for (int t = 0; t < 4; ++t) {
      int c0 = (k0 + khi + 0) * H + hb + 16 * t + lrow;
      int c1 = (k0 + khi + 1) * H + hb + 16 * t + lrow;
      v2f b3; b3.x = Wv[c0];       b3.y = Wv[c1];
      f[t] = wmma_f32(a3, b3, f[t]);
      v2f b1; b1.x = Wq_first[c0]; b1.y = Wq_first[c1];
      f[t] = wmma_f32(a1, b1, f[t]);
      v2f b2; b2.x = Wq_last[c0];  b2.y = Wq_last[c1];
      f[t] = wmma_f32(a2, b2, f[t]);
    }
  }

  // epilogue: + q_graph + load*W_load + b_load ; store final_q
#pragma unroll
  for (int t = 0; t < 4; ++t) {
    int h = hb + 16 * t + lrow;
    float wl = W_load[h], bl = b_load[h], qg = s_qg[h];
#pragma unroll
    for (int r = 0; r < 8; ++r) {
      int p = p0 + r + 8 * (lane >> 4);
      fq[((size_t)b * P + p) * H + h] = f[t][r] + qg + s_load[p] * wl + bl;
    }
  }
}

// ---------------------------------------------------------------------------
// Kernel 2: logits[b,p,n] = 10*tanh(fq.emb/sqrt(H) - dist/sqrt(2)) + maskprob
// Two workgroups per batch (even/odd n-tiles); each wave does one n-tile at a
// time against ALL four p-tiles so each B fetch feeds 4 WMMAs.
// ---------------------------------------------------------------------------
__global__ void __launch_bounds__(256)
score_kernel(const float* __restrict__ dists, const float* __restrict__ emb,
             const int* __restrict__ last_node, const float* __restrict__ gmask,
             const float* __restrict__ fq, float* __restrict__ out)
{
  __shared__ __align__(16) float s_fq[64 * EMB_STR];  // 33.8 KB
  __shared__ int s_ln[64];

  const int b      = blockIdx.x >> 1;
  const int parity = blockIdx.x & 1;
  const int tid    = threadIdx.x;
  const int lane   = tid & 31;
  const int wid    = tid >> 5;
  const int lrow   = lane & 15;
  const int khi    = (lane >> 4) * 2;

  const float* fqB = fq + (size_t)b * P * H;
  for (int i = tid; i < 64 * 32; i += 256) {     // stage final_q[b] (async)
    int r = i >> 5, c4 = (i & 31) * 4;
    copy16_g2l(&s_fq[r * EMB_STR + c4], fqB + (size_t)r * H + c4);
  }
  if (tid < 64) s_ln[tid] = last_node[b * P + tid];
  wait_async_copies();
  __syncthreads();

  const float* embB = emb + (size_t)b * NN * H;
  const float inv_sqrtH = 0.08838834764831845f;  // 1/sqrt(128)
  const float inv_sqrt2 = 0.7071067811865476f;

  for (int t = parity + 2 * wid; t < (NN + 15) / 16; t += 16) {
    const int n0 = t * 16;
    const int n  = n0 + lrow;
    const int nc = (n < NN) ? n : (NN - 1);      // clamp address, no EXEC split
    const v2f* embrow = (const v2f*)(embB + (size_t)nc * H);
    v8f acc[4] = {};
    for (int k0 = 0; k0 < H; k0 += 4) {
      v2f bb = embrow[(k0 >> 1) + (lane >> 4)];  // B pair: one b64 load
#pragma unroll
      for (int q = 0; q < 4; ++q) {
        v2f a = *(const v2f*)&s_fq[(q * 16 + lrow) * EMB_STR + k0 + khi];
        acc[q] = wmma_f32(a, bb, acc[q]);
      }
    }
#pragma unroll
    for (int q = 0; q < 4; ++q) {
#pragma unroll
      for (int r = 0; r < 8; ++r) {
        if (n < NN) {
          int p = q * 16 + r + 8 * (lane >> 4);
          float d  = dists[((size_t)b * NN + s_ln[p]) * NN + n];
          float s  = acc[q][r] * inv_sqrtH - d * inv_sqrt2;
          float sc = 10.f * tanhf(s);
          float g  = gmask[((size_t)b * P + p) * NN + n];
          out[((size_t)b * P + p) * NN + n] = sc + ((g < -1e30f) ? -1e8f : g);
        }
      }
    }
  }
}

// ---------------------------------------------------------------------------
// Kernel 3: in-place row softmax over N (exp cached in LDS: 2 global passes)
// ---------------------------------------------------------------------------
__global__ void __launch_bounds__(256)
softmax_kernel(float* __restrict__ out)
{
  __shared__ float red[256];
  __shared__ float s_e[NN];
  float* row = out + (size_t)blockIdx.x * NN;
  const int tid = threadIdx.x;

  float m = -3.4e38f;
  for (int i = tid; i < NN; i += 256) m = fmaxf(m, row[i]);
  red[tid] = m; __syncthreads();
  for (int s = 128; s > 0; s >>= 1) {
    if (tid < s) red[tid] = fmaxf(red[tid], red[tid + s]);
    __syncthreads();
  }
  m = red[0]; __syncthreads();

  float sum = 0.f;
  for (int i = tid; i < NN; i += 256) {
    float e = expf(row[i] - m);
    s_e[i] = e;
    sum += e;
  }
  red[tid] = sum; __syncthreads();
  for (int s = 128; s > 0; s >>= 1) {
    if (tid < s) red[tid] += red[tid + s];
    __syncthreads();
  }
  float inv = 1.0f / red[0];

  for (int i = tid; i < NN; i += 256) row[i] = s_e[i] * inv;
}

// ---------------------------------------------------------------------------
extern "C" void kernel_launch(void* const* d_in, const int* in_sizes, int n_in,
                              void* d_out, int out_size, void* d_ws, size_t ws_size,
                              hipStream_t stream) {
  const float* dists     = (const float*)d_in[0];
  const float* emb       = (const float*)d_in[1];
  const float* encq1     = (const float*)d_in[2];
  const int*   last_node = (const int*)d_in[3];
  const float* loadv     = (const float*)d_in[4];
  const float* gmask     = (const float*)d_in[5];
  const float* Wq_graph  = (const float*)d_in[6];
  const float* Wq_first  = (const float*)d_in[7];
  const float* Wq_last   = (const float*)d_in[8];
  const float* Wv        = (const float*)d_in[9];
  const float* W_load    = (const float*)d_in[10];
  const float* b_load    = (const float*)d_in[11];
  float* out = (float*)d_out;
  float* fq  = (float*)d_ws;   // B*P*H floats = 2 MB scratch

  fq_kernel<<<B, 256, 0, stream>>>(emb, encq1, last_node, loadv, gmask,
                                   Wq_graph, Wq_first, Wq_last, Wv,
                                   W_load, b_load, fq);
  score_kernel<<<B * 2, 256, 0, stream>>>(dists, emb, last_node, gmask, fq, out);
  softmax_kernel<<<B * P, 256, 0, stream>>>(out);
}